// MoEBlock_33071248179926
// MI455X (gfx1250) — compile-verified
//
#include <hip/hip_runtime.h>

// ---------------- problem constants (from reference) ----------------
#define T_TOK 8192   // B*S tokens
#define D_DIM 2048
#define F_DIM 8192
#define E_NUM 8
#define K_TOP 2

// ---------------- GEMM tiling ----------------
#define BM 128
#define BN 128
#define BK 32
#define LDT (BK + 2)   // LDS stride (floats): even -> 8B-aligned v2f, 17*2 bank spread

typedef __attribute__((ext_vector_type(2))) float v2f;
typedef __attribute__((ext_vector_type(8))) float v8f;

// D = A(16x4 f32) * B(4x16 f32) + C(16x16 f32), wave32, CDNA5 WMMA
__device__ __forceinline__ v8f wmma4(v2f a, v2f b, v8f c) {
  return __builtin_amdgcn_wmma_f32_16x16x4_f32(false, a, false, b, (short)0, c,
                                               false, false);
}

__device__ __forceinline__ float bf2f(unsigned short h) {
  unsigned int u = ((unsigned int)h) << 16;
  float f;
  __builtin_memcpy(&f, &u, 4);
  return f;
}
__device__ __forceinline__ unsigned short f2bf(float f) {
  unsigned int u;
  __builtin_memcpy(&u, &f, 4);
  u += 0x7FFFu + ((u >> 16) & 1u);   // round-to-nearest-even
  return (unsigned short)(u >> 16);
}

// -------- CDNA5 async global->LDS copy (ASYNCcnt-tracked), with fallback ----
#if defined(__gfx1250__) && __has_builtin(__builtin_amdgcn_global_load_async_to_lds_b32)
#define USE_ASYNC_LDS 1
#else
#define USE_ASYNC_LDS 0
#endif

__device__ __forceinline__ void async_wait0() {
#if __has_builtin(__builtin_amdgcn_s_wait_asynccnt)
  __builtin_amdgcn_s_wait_asynccnt(0);
#else
  asm volatile("s_wait_asynccnt 0x0" ::: "memory");
#endif
}

#if USE_ASYNC_LDS
typedef __attribute__((address_space(1))) int gas_i32;  // global
typedef __attribute__((address_space(3))) int las_i32;  // LDS
// per-lane 4B global->LDS DMA: global side coalesced, LDS side may scatter
__device__ __forceinline__ void async_copy4(const float* gsrc, float* ldst) {
  __builtin_amdgcn_global_load_async_to_lds_b32((gas_i32*)gsrc, (las_i32*)ldst,
                                                0, 0);
}
#endif

// ---------------- workspace layout (bytes) ----------------
#define OFF_COUNTS   0
#define OFF_ROWBASE  256
#define OFF_TOKLIST  512
#define OFF_TKE      262656
#define OFF_TKSLOT   328192
#define OFF_TKCW     393728
#define OFF_H        1048576      // bf16 h[T*K][F]  (256 MiB)
#define OFF_Y        269484032ULL // f32  y[T*K][D]  (128 MiB)

// ============================================================
// 0) zero per-expert counters (ws is poisoned; re-init every launch)
// ============================================================
__global__ void moe_init_kernel(int* counts) {
  if (threadIdx.x < E_NUM) counts[threadIdx.x] = 0;
}

// ============================================================
// 1) router: one wave per token; top-2 + softmax; claim compacted slots
// ============================================================
__global__ __launch_bounds__(256) void moe_router_kernel(
    const float* __restrict__ x, const float* __restrict__ rw, int* counts,
    int* tokList, int* tkE, int* tkSlot, float* tkCw) {
  const int lane = threadIdx.x & 31;
  const int t = blockIdx.x * 8 + (threadIdx.x >> 5);

  float acc[E_NUM];
#pragma unroll
  for (int e = 0; e < E_NUM; ++e) acc[e] = 0.f;

  const float* xr = x + (size_t)t * D_DIM;
  for (int d = lane; d < D_DIM; d += 32) {
    float xv = xr[d];
    const float* r = rw + (size_t)d * E_NUM;
#pragma unroll
    for (int e = 0; e < E_NUM; ++e) acc[e] += xv * r[e];
  }
#pragma unroll
  for (int off = 16; off >= 1; off >>= 1) {
#pragma unroll
    for (int e = 0; e < E_NUM; ++e) acc[e] += __shfl_xor(acc[e], off, 32);
  }

  if (lane == 0) {
    int e0 = 0, e1 = -1;
    float l0 = acc[0], l1 = -3.0e38f;
#pragma unroll
    for (int e = 1; e < E_NUM; ++e) {
      float v = acc[e];
      if (v > l0) { l1 = l0; e1 = e0; l0 = v; e0 = e; }
      else if (v > l1) { l1 = v; e1 = e; }
    }
    float w0 = 1.f / (1.f + expf(l1 - l0));  // softmax over the two logits
    float w1 = 1.f - w0;
    int s0 = atomicAdd(&counts[e0], 1);
    int s1 = atomicAdd(&counts[e1], 1);
    tokList[e0 * T_TOK + s0] = t;
    tokList[e1 * T_TOK + s1] = t;
    tkE[2 * t] = e0;     tkE[2 * t + 1] = e1;
    tkSlot[2 * t] = s0;  tkSlot[2 * t + 1] = s1;
    tkCw[2 * t] = w0;    tkCw[2 * t + 1] = w1;
  }
}

// ============================================================
// 2) exclusive prefix over 8 expert counts
// ============================================================
__global__ void moe_rowbase_kernel(const int* counts, int* rowBase) {
  if (threadIdx.x == 0) {
    int s = 0;
    for (int e = 0; e < E_NUM; ++e) { rowBase[e] = s; s += counts[e]; }
  }
}

// ============================================================
// shared GEMM core pieces
//   As[m][k] stride LDT ; Bs[n][k] stride LDT (transposed)
//   -> every WMMA fragment is ONE aligned ds_load_b64 (k, k+1 adjacent)
//   B fill: per-lane async b32 (global coalesced, LDS transpose-scatter)
// ============================================================
#if USE_ASYNC_LDS
#define B_FILL_ONE(wp, NSTRIDE, k, n)                                          \
  async_copy4(wp + (size_t)(k) * (NSTRIDE) + (n), &Bs[(n) * LDT + (k)]);
#define A_FILL_ONE(gp, r, c) async_copy4((gp), &As[(r) * LDT + (c)]);
#else
#define B_FILL_ONE(wp, NSTRIDE, k, n)                                          \
  Bs[(n) * LDT + (k)] = wp[(size_t)(k) * (NSTRIDE) + (n)];
#define A_FILL_ONE(gp, r, c) As[(r) * LDT + (c)] = *(gp);
#endif

#define GEMM_FILL_B(wp, NSTRIDE)                                               \
  {                                                                            \
    _Pragma("unroll") for (int i = 0; i < 16; ++i) {                           \
      int k = (tid >> 7) + 2 * i; /* 32 k-rows x 128 n */                      \
      int n = tid & 127;          /* lane-consecutive n: coalesced */          \
      B_FILL_ONE(wp, NSTRIDE, k, n)                                            \
    }                                                                          \
  }

#define GEMM_COMPUTE_BK()                                                      \
  _Pragma("unroll") for (int ks = 0; ks < BK / 4; ++ks) {                      \
    int kk = ks * 4 + 2 * lh; /* lanes 16-31 own K+2..K+3 */                   \
    v2f a0 = *(const v2f*)&As[(mo + ll) * LDT + kk];                           \
    v2f a1 = *(const v2f*)&As[(mo + 16 + ll) * LDT + kk];                      \
    v2f bf0 = *(const v2f*)&Bs[(no + ll) * LDT + kk];                          \
    v2f bf1 = *(const v2f*)&Bs[(no + 16 + ll) * LDT + kk];                     \
    v2f bf2 = *(const v2f*)&Bs[(no + 32 + ll) * LDT + kk];                     \
    v2f bf3 = *(const v2f*)&Bs[(no + 48 + ll) * LDT + kk];                     \
    acc[0][0] = wmma4(a0, bf0, acc[0][0]);                                     \
    acc[0][1] = wmma4(a0, bf1, acc[0][1]);                                     \
    acc[0][2] = wmma4(a0, bf2, acc[0][2]);                                     \
    acc[0][3] = wmma4(a0, bf3, acc[0][3]);                                     \
    acc[1][0] = wmma4(a1, bf0, acc[1][0]);                                     \
    acc[1][1] = wmma4(a1, bf1, acc[1][1]);                                     \
    acc[1][2] = wmma4(a1, bf2, acc[1][2]);                                     \
    acc[1][3] = wmma4(a1, bf3, acc[1][3]);                                     \
  }

// ============================================================
// 3) GEMM1: h[row] = gelu(x[tok(row)] @ w1[e] + b1[e]), store bf16
//    grid = (F/BN, T/BM, E); 8 waves; wave computes a 32x64 patch
//    A tile is async too: out-of-range slots clamp to a valid token row
//    (garbage rows computed but never stored: epilogue guards s < cnt)
// ============================================================
__global__ __launch_bounds__(256) void moe_gemm1_kernel(
    const float* __restrict__ x, const float* __restrict__ w1,
    const float* __restrict__ bias1, const int* __restrict__ counts,
    const int* __restrict__ rowBase, const int* __restrict__ tokList,
    unsigned short* __restrict__ hbuf) {
  const int e = blockIdx.z;
  const int cnt = counts[e];
  const int m0 = blockIdx.y * BM;
  if (m0 >= cnt) return;                       // uniform: EXEC stays all-1s
  const int n0 = blockIdx.x * BN;
  const int rbase = rowBase[e];

  __shared__ float As[BM * LDT];
  __shared__ float Bs[BN * LDT];

  const int tid = threadIdx.x;
  const int lane = tid & 31, wave = tid >> 5;
  const int ll = lane & 15, lh = lane >> 4;
  const int mo = (wave & 3) * 32, no = (wave >> 2) * 64;

  const int ac = tid & 31;                     // A load: k column
  int atok[16];
#pragma unroll
  for (int i = 0; i < 16; ++i) {               // gathered token per A row
    int s = m0 + (tid >> 5) + 8 * i;
    atok[i] = tokList[e * T_TOK + (s < cnt ? s : cnt - 1)];  // clamp, not zero
  }

  v8f acc[2][4];
#pragma unroll
  for (int i = 0; i < 2; ++i)
#pragma unroll
    for (int j = 0; j < 4; ++j) acc[i][j] = {};

  for (int d0 = 0; d0 < D_DIM; d0 += BK) {
    __syncthreads();
    const float* wp = w1 + ((size_t)e * D_DIM + d0) * F_DIM + n0;
    GEMM_FILL_B(wp, F_DIM)
#pragma unroll
    for (int i = 0; i < 16; ++i) {
      int r = (tid >> 5) + 8 * i;
      A_FILL_ONE(x + (size_t)atok[i] * D_DIM + d0 + ac, r, ac)
    }
    async_wait0();
    __syncthreads();
    GEMM_COMPUTE_BK()
  }

  const float* bp = bias1 + (size_t)e * F_DIM;
#pragma unroll
  for (int j = 0; j < 4; ++j) {
    int n = n0 + no + j * 16 + ll;
    float bv = bp[n];
#pragma unroll
    for (int i = 0; i < 2; ++i) {
#pragma unroll
      for (int r = 0; r < 8; ++r) {            // C: vgpr r -> M = r + 8*lh
        int s = m0 + mo + i * 16 + 8 * lh + r;
        if (s < cnt) {
          float v = acc[i][j][r] + bv;
          float g = 0.5f * v * (1.f + erff(v * 0.70710678118654752f));
          hbuf[(size_t)(rbase + s) * F_DIM + n] = f2bf(g);
        }
      }
    }
  }
}

// ============================================================
// 4) GEMM2: y[row] = h[row] @ w2[e] + b2[e]   (h bf16 -> f32 at LDS fill;
//    A fill stays manual because of the dequant)
//    grid = (D/BN, T/BM, E)
// ============================================================
__global__ __launch_bounds__(256) void moe_gemm2_kernel(
    const unsigned short* __restrict__ hbuf, const float* __restrict__ w2,
    const float* __restrict__ bias2, const int* __restrict__ counts,
    const int* __restrict__ rowBase, float* __restrict__ ybuf) {
  const int e = blockIdx.z;
  const int cnt = counts[e];
  const int m0 = blockIdx.y * BM;
  if (m0 >= cnt) return;
  const int n0 = blockIdx.x * BN;
  const int rbase = rowBase[e];

  __shared__ float As[BM * LDT];
  __shared__ float Bs[BN * LDT];

  const int tid = threadIdx.x;
  const int lane = tid & 31, wave = tid >> 5;
  const int ll = lane & 15, lh = lane >> 4;
  const int mo = (wave & 3) * 32, no = (wave >> 2) * 64;

  const int ac = tid & 31;
  int hrow[16];
#pragma unroll
  for (int i = 0; i < 16; ++i) {
    int s = m0 + (tid >> 5) + 8 * i;
    hrow[i] = rbase + (s < cnt ? s : cnt - 1);  // clamp to a written row
  }

  v8f acc[2][4];
#pragma unroll
  for (int i = 0; i < 2; ++i)
#pragma unroll
    for (int j = 0; j < 4; ++j) acc[i][j] = {};

  for (int f0 = 0; f0 < F_DIM; f0 += BK) {
    __syncthreads();
    const float* wp = w2 + ((size_t)e * F_DIM + f0) * D_DIM + n0;
    GEMM_FILL_B(wp, D_DIM)
#pragma unroll
    for (int i = 0; i < 16; ++i) {
      int r = (tid >> 5) + 8 * i;
      As[r * LDT + ac] = bf2f(hbuf[(size_t)hrow[i] * F_DIM + f0 + ac]);
    }
    async_wait0();
    __syncthreads();
    GEMM_COMPUTE_BK()
  }

  const float* bp = bias2 + (size_t)e * D_DIM;
#pragma unroll
  for (int j = 0; j < 4; ++j) {
    int n = n0 + no + j * 16 + ll;
    float bv = bp[n];
#pragma unroll
    for (int i = 0; i < 2; ++i) {
#pragma unroll
      for (int r = 0; r < 8; ++r) {
        int s = m0 + mo + i * 16 + 8 * lh + r;
        if (s < cnt)
          ybuf[(size_t)(rbase + s) * D_DIM + n] = acc[i][j][r] + bv;
      }
    }
  }
}

// ============================================================
// 5) combine: out[t] = x[t] + cw0*y[row0] + cw1*y[row1]  (residual fused)
// ============================================================
__global__ __launch_bounds__(256) void moe_combine_kernel(
    const float* __restrict__ x, const float* __restrict__ ybuf,
    const int* __restrict__ rowBase, const int* __restrict__ tkE,
    const int* __restrict__ tkSlot, const float* __restrict__ tkCw,
    float* __restrict__ out) {
  const int t = blockIdx.x;
  const int e0 = tkE[2 * t], e1 = tkE[2 * t + 1];
  const size_t r0 = (size_t)(rowBase[e0] + tkSlot[2 * t]);
  const size_t r1 = (size_t)(rowBase[e1] + tkSlot[2 * t + 1]);
  const float w0 = tkCw[2 * t], w1 = tkCw[2 * t + 1];
  const float4* xp = (const float4*)(x + (size_t)t * D_DIM);
  const float4* y0 = (const float4*)(ybuf + r0 * D_DIM);
  const float4* y1 = (const float4*)(ybuf + r1 * D_DIM);
  float4* op = (float4*)(out + (size_t)t * D_DIM);
  for (int i = threadIdx.x; i < D_DIM / 4; i += blockDim.x) {
    float4 a = xp[i], p = y0[i], q = y1[i], o;
    o.x = a.x + w0 * p.x + w1 * q.x;
    o.y = a.y + w0 * p.y + w1 * q.y;
    o.z = a.z + w0 * p.z + w1 * q.z;
    o.w = a.w + w0 * p.w + w1 * q.w;
    op[i] = o;
  }
}

// ============================================================
extern "C" void kernel_launch(void* const* d_in, const int* in_sizes, int n_in,
                              void* d_out, int out_size, void* d_ws,
                              size_t ws_size, hipStream_t stream) {
  (void)in_sizes; (void)n_in; (void)out_size; (void)ws_size;
  const float* x  = (const float*)d_in[0];
  const float* rw = (const float*)d_in[1];
  const float* w1 = (const float*)d_in[2];
  const float* b1 = (const float*)d_in[3];
  const float* w2 = (const float*)d_in[4];
  const float* b2 = (const float*)d_in[5];
  float* out = (float*)d_out;

  char* ws = (char*)d_ws;
  int*   counts  = (int*)(ws + OFF_COUNTS);
  int*   rowBase = (int*)(ws + OFF_ROWBASE);
  int*   tokList = (int*)(ws + OFF_TOKLIST);
  int*   tkE     = (int*)(ws + OFF_TKE);
  int*   tkSlot  = (int*)(ws + OFF_TKSLOT);
  float* tkCw    = (float*)(ws + OFF_TKCW);
  unsigned short* hbuf = (unsigned short*)(ws + OFF_H);
  float* ybuf    = (float*)(ws + OFF_Y);

  moe_init_kernel<<<1, 32, 0, stream>>>(counts);
  moe_router_kernel<<<T_TOK / 8, 256, 0, stream>>>(x, rw, counts, tokList, tkE,
                                                   tkSlot, tkCw);
  moe_rowbase_kernel<<<1, 32, 0, stream>>>(counts, rowBase);

  dim3 g1(F_DIM / BN, T_TOK / BM, E_NUM);
  moe_gemm1_kernel<<<g1, 256, 0, stream>>>(x, w1, b1, counts, rowBase, tokList,
                                           hbuf);
  dim3 g2(D_DIM / BN, T_TOK / BM, E_NUM);
  moe_gemm2_kernel<<<g2, 256, 0, stream>>>(hbuf, w2, b2, counts, rowBase, ybuf);

  moe_combine_kernel<<<T_TOK, 256, 0, stream>>>(x, ybuf, rowBase, tkE, tkSlot,
                                                tkCw, out);
}